// pendelum1dof_43379169689809
// MI455X (gfx1250) — compile-verified
//
#include <hip/hip_runtime.h>
#include <hip/hip_bf16.h>

// Pendulum 1-DOF Hamiltonian dynamics, closed form:
//   out[i,0] = y[i,1]                    (dq = p)
//   out[i,1] = -9.81f * sinf(y[i,0])     (dp = -G*sin(theta)), M=L=1
//
// Memory-bound elementwise map: 128 MiB total traffic -> ~5.5 us at 23.3 TB/s.
// Strategy: 128-bit vector loads/stores (two (theta,p) rows per float4),
// wave32-friendly 256-thread blocks, grid-stride loop for full coverage.

__global__ void pendulum_dyn_kernel(const float4* __restrict__ y4,
                                    float4* __restrict__ out4,
                                    int n4) {
    const float NEG_G = -9.81f;
    int idx    = blockIdx.x * blockDim.x + threadIdx.x;
    int stride = gridDim.x * blockDim.x;
    for (int i = idx; i < n4; i += stride) {
        // One b128 load covers rows 2i (v.x=theta0, v.y=p0) and 2i+1 (v.z, v.w).
        float4 v = y4[i];
        float4 r;
        r.x = v.y;                  // dq0 = p0
        r.y = NEG_G * sinf(v.x);    // dp0 = -G sin(theta0)
        r.z = v.w;                  // dq1 = p1
        r.w = NEG_G * sinf(v.z);    // dp1 = -G sin(theta1)
        out4[i] = r;                // one b128 store
    }
}

// Scalar tail kernel in case total element count isn't a multiple of 4
// (not hit for B = 2^23, but keeps kernel_launch fully general).
__global__ void pendulum_dyn_tail(const float* __restrict__ y,
                                  float* __restrict__ out,
                                  int start_row, int n_rows) {
    int r = start_row + blockIdx.x * blockDim.x + threadIdx.x;
    if (r >= n_rows) return;
    float theta = y[2 * r + 0];
    float p     = y[2 * r + 1];
    out[2 * r + 0] = p;
    out[2 * r + 1] = -9.81f * sinf(theta);
}

extern "C" void kernel_launch(void* const* d_in, const int* in_sizes, int n_in,
                              void* d_out, int out_size, void* d_ws, size_t ws_size,
                              hipStream_t stream) {
    // d_in[0] = t (unused, 1 float), d_in[1] = y, flat [B,2] float32.
    const float* y   = (const float*)d_in[1];
    float*       out = (float*)d_out;

    const int n_floats = in_sizes[1];       // B * 2
    const int n4       = n_floats / 4;      // float4 chunks (2 rows each)

    if (n4 > 0) {
        const int block = 256;              // 8 wave32 waves per block
        long long want  = ((long long)n4 + block - 1) / block;
        int grid = (int)(want > 1048576 ? 1048576 : want);  // grid-stride if clamped
        pendulum_dyn_kernel<<<grid, block, 0, stream>>>(
            (const float4*)y, (float4*)out, n4);
    }

    // Handle any rows not covered by the float4 path (n_floats % 4 != 0).
    const int rows_done = (n4 * 4) / 2;
    const int n_rows    = n_floats / 2;
    if (rows_done < n_rows) {
        int rem = n_rows - rows_done;
        pendulum_dyn_tail<<<(rem + 255) / 256, 256, 0, stream>>>(
            y, out, rows_done, n_rows);
    }
}